// CRF1_3573412790839
// MI455X (gfx1250) — compile-verified
//
#include <hip/hip_runtime.h>

#define Bn 128
#define Ln 1024
#define Tn 128
#define PD 8   // async prefetch depth (outstanding global_load_async_to_lds per wave)

typedef float v4f __attribute__((ext_vector_type(4)));

// ---------------- fast math helpers (single v_exp_f32 / v_log_f32) ----------
__device__ __forceinline__ float fast_exp2(float x) {
#if __has_builtin(__builtin_amdgcn_exp2f)
  return __builtin_amdgcn_exp2f(x);
#else
  return exp2f(x);
#endif
}
__device__ __forceinline__ float fast_log2(float x) {
#if __has_builtin(__builtin_amdgcn_logf)
  return __builtin_amdgcn_logf(x);   // hardware V_LOG_F32 == log2
#else
  return log2f(x);
#endif
}

// ---------------- CDNA5 async global->LDS copy (ASYNCcnt path) --------------
__device__ __forceinline__ void async_g2l_b32(const float* g, float* l) {
#if __has_builtin(__builtin_amdgcn_global_load_async_to_lds_b32)
  __builtin_amdgcn_global_load_async_to_lds_b32(
      (__attribute__((address_space(1))) int*)g,
      (__attribute__((address_space(3))) int*)l,
      0, 0);
#else
  unsigned lofs = (unsigned)(__UINTPTR_TYPE__)(__attribute__((address_space(3))) int*)l;
  asm volatile("global_load_async_to_lds_b32 %0, %1, off"
               :: "v"(lofs), "v"(g) : "memory");
#endif
}

template <int N>
__device__ __forceinline__ void wait_async_le() {
#if __has_builtin(__builtin_amdgcn_s_wait_asynccnt)
  __builtin_amdgcn_s_wait_asynccnt(N);
#else
  asm volatile("s_wait_asynccnt %0" :: "n"(N) : "memory");
#endif
}

// ---------------- kernel 1: partial sums of x -------------------------------
__global__ __launch_bounds__(256) void sumx_kernel(const float* __restrict__ x,
                                                   float* __restrict__ ws) {
  const size_t N4 = ((size_t)Bn * Ln * Tn) / 4;
  size_t idx = (size_t)blockIdx.x * 256 + threadIdx.x;
  float s = 0.f;
  const v4f* x4 = (const v4f*)x;
  for (size_t k = idx; k < N4; k += (size_t)256 * 256) {
    v4f v = __builtin_nontemporal_load(&x4[k]);
    s += (v.x + v.y) + (v.z + v.w);
  }
  s += __shfl_xor(s, 16, 32);
  s += __shfl_xor(s, 8, 32);
  s += __shfl_xor(s, 4, 32);
  s += __shfl_xor(s, 2, 32);
  s += __shfl_xor(s, 1, 32);
  __shared__ float sred[8];
  if ((threadIdx.x & 31) == 0) sred[threadIdx.x >> 5] = s;
  __syncthreads();
  if (threadIdx.x == 0) {
    float t = 0.f;
#pragma unroll
    for (int q = 0; q < 8; ++q) t += sred[q];
    ws[blockIdx.x] = t;
  }
}

// ---------------- kernel 2: CRF forward scan, one workgroup per batch -------
__global__ __launch_bounds__(512, 1) void crf_forward_kernel(
    const float* __restrict__ x, const float* __restrict__ trans,
    float* __restrict__ out_mx, float* __restrict__ out_pre,
    float* __restrict__ lse_ws) {
  extern __shared__ char smem_raw[];
  float2* te = (float2*)smem_raw;          // [Tn*Tn] (trans, exp(trans)) 128 KB

  __shared__ float2 se[Tn];                // (s_i, exp(s_i - m_s))
  __shared__ float prevs[Tn];
  __shared__ float obs[PD][Tn];            // async prefetch ring, 8 deep (4 KB)
  __shared__ float pm[4][Tn];
  __shared__ int pidx[4][Tn];
  __shared__ float pf[4][Tn];
  __shared__ float red[32];

  const int tid = threadIdx.x;
  const int j = tid & (Tn - 1);
  const int iq = tid >> 7;                 // i-segment 0..3
  const int wv = tid >> 5;                 // wave 0..15
  const int ln = tid & 31;
  const int b = blockIdx.x;

  const float* xb = x + (size_t)b * Ln * Tn;
  const float LOG2E = 1.4426950408889634f;
  const float LN2 = 0.6931471805599453f;

  // Stage transitions (value, exp(value)) pairs into LDS (conflict-free b64).
  for (int k = tid; k < Tn * Tn; k += 512) {
    float tr = trans[k];
    te[k] = make_float2(tr, fast_exp2(tr * LOG2E));
  }

  // t = 0 row: prev0 = x[b,0,:] + transitions[0,:]; argmax row = iota.
  // Prime the async pipeline with obs for t = 1..PD-1 (7 outstanding).
  if (tid < Tn) {
    float p0 = xb[j] + trans[j];
    prevs[j] = p0;
    __builtin_nontemporal_store(p0, &out_mx[(size_t)b * Ln * Tn + j]);
    __builtin_nontemporal_store((float)j, &out_pre[(size_t)b * Ln * Tn + j]);
#pragma unroll
    for (int tt = 1; tt < PD; ++tt)
      async_g2l_b32(xb + (size_t)tt * Tn + j, &obs[tt & (PD - 1)][j]);
  }

  for (int t = 1; t < Ln; ++t) {
    float m_s = 0.f, s_j = 0.f;
    if (tid < Tn) {
      // issue prefetch for t+PD-1 (clamped; keeps ASYNCcnt uniform: 8 in flight)
      const int tn = (t + PD - 1 < Ln) ? (t + PD - 1) : (Ln - 1);
      async_g2l_b32(xb + (size_t)tn * Tn + j, &obs[(t + PD - 1) & (PD - 1)][j]);
      wait_async_le<PD - 1>();             // oldest (this step's obs) done
      s_j = prevs[j] + obs[t & (PD - 1)][j];
      float v = s_j;                       // wave max for numerical shift
      v = fmaxf(v, __shfl_xor(v, 16, 32));
      v = fmaxf(v, __shfl_xor(v, 8, 32));
      v = fmaxf(v, __shfl_xor(v, 4, 32));
      v = fmaxf(v, __shfl_xor(v, 2, 32));
      v = fmaxf(v, __shfl_xor(v, 1, 32));
      if (ln == 0) red[wv] = v;
    }
    __syncthreads();                       // B1
    if (tid < Tn) {
      m_s = fmaxf(fmaxf(red[0], red[1]), fmaxf(red[2], red[3]));
      se[j] = make_float2(s_j, fast_exp2((s_j - m_s) * LOG2E));
    }
    __syncthreads();                       // B2

    // max-plus (max/argmax over i) + exp-domain matvec (logsumexp), i-segment
    float mj = -__builtin_inff();
    int aj = 0;
    float fj = 0.f;
    const float2* terow = te + (size_t)iq * 32 * Tn + j;
#pragma unroll 8
    for (int k = 0; k < 32; ++k) {
      const int i = iq * 32 + k;
      float2 sp = se[i];                   // broadcast ds_load_b64
      float2 tp = terow[(size_t)k * Tn];   // conflict-free ds_load_b64
      float v = sp.x + tp.x;
      if (v > mj) { mj = v; aj = i; }      // strict > keeps first max (ties)
      fj = __builtin_fmaf(sp.y, tp.y, fj);
    }
    pm[iq][j] = mj;
    pidx[iq][j] = aj;
    pf[iq][j] = fj;
    __syncthreads();                       // B3
    if (tid < Tn) {
      float bm = pm[0][j];
      int ba = pidx[0][j];
      float fs = pf[0][j];
#pragma unroll
      for (int q = 1; q < 4; ++q) {
        float qm = pm[q][j];
        if (qm > bm) { bm = qm; ba = pidx[q][j]; }
        fs += pf[q][j];
      }
      prevs[j] = m_s + LN2 * fast_log2(fs);       // new alpha
      const size_t o = ((size_t)b * Ln + t) * Tn + j;
      __builtin_nontemporal_store(bm, &out_mx[o]);
      __builtin_nontemporal_store((float)ba, &out_pre[o]);
    }
    // no extra barrier: B1 of the next iteration orders pm/pf/red reuse
  }

  // final log_z contribution: logsumexp(prev_final[b,:])
  float v = (tid < Tn) ? prevs[j] : -__builtin_inff();
  {
    float r = v;
    r = fmaxf(r, __shfl_xor(r, 16, 32));
    r = fmaxf(r, __shfl_xor(r, 8, 32));
    r = fmaxf(r, __shfl_xor(r, 4, 32));
    r = fmaxf(r, __shfl_xor(r, 2, 32));
    r = fmaxf(r, __shfl_xor(r, 1, 32));
    if (ln == 0) red[wv] = r;
  }
  __syncthreads();
  float mfin = red[0];
#pragma unroll
  for (int q = 1; q < 16; ++q) mfin = fmaxf(mfin, red[q]);
  float e = (tid < Tn) ? fast_exp2((v - mfin) * LOG2E) : 0.f;
  e += __shfl_xor(e, 16, 32);
  e += __shfl_xor(e, 8, 32);
  e += __shfl_xor(e, 4, 32);
  e += __shfl_xor(e, 2, 32);
  e += __shfl_xor(e, 1, 32);
  if (ln == 0) red[16 + wv] = e;
  __syncthreads();
  if (tid == 0) {
    float fs = 0.f;
#pragma unroll
    for (int q = 0; q < 16; ++q) fs += red[16 + q];
    lse_ws[b] = mfin + LN2 * fast_log2(fs);
  }
}

// ---------------- kernel 3: likelihood = sum(x)+sum(trans) - sum(lse_b) -----
__global__ __launch_bounds__(256) void finalize_kernel(
    const float* __restrict__ trans, const float* __restrict__ ws_x,
    const float* __restrict__ lse_ws, float* __restrict__ out0) {
  const int tid = threadIdx.x;
  float s = ws_x[tid];                      // 256 x-partials
  for (int k = tid; k < Tn * Tn; k += 256) s += trans[k];
  if (tid < Bn) s -= lse_ws[tid];
  s += __shfl_xor(s, 16, 32);
  s += __shfl_xor(s, 8, 32);
  s += __shfl_xor(s, 4, 32);
  s += __shfl_xor(s, 2, 32);
  s += __shfl_xor(s, 1, 32);
  __shared__ float sred[8];
  if ((tid & 31) == 0) sred[tid >> 5] = s;
  __syncthreads();
  if (tid == 0) {
    float t = 0.f;
#pragma unroll
    for (int q = 0; q < 8; ++q) t += sred[q];
    out0[0] = t;
  }
}

// ---------------- host launcher ---------------------------------------------
extern "C" void kernel_launch(void* const* d_in, const int* in_sizes, int n_in,
                              void* d_out, int out_size, void* d_ws,
                              size_t ws_size, hipStream_t stream) {
  (void)in_sizes; (void)n_in; (void)out_size; (void)ws_size;
  const float* x = (const float*)d_in[0];
  const float* trans = (const float*)d_in[1];

  float* out = (float*)d_out;
  float* out_mx = out + 1;                               // max_score [B,L,T]
  float* out_pre = out + 1 + (size_t)Bn * Ln * Tn;       // argmax    [B,L,T]

  float* ws_x = (float*)d_ws;                            // 256 partials
  float* lse_ws = ws_x + 256;                            // 128 per-batch lse

  sumx_kernel<<<256, 256, 0, stream>>>(x, ws_x);

  const size_t lds_bytes = (size_t)Tn * Tn * sizeof(float2);  // 128 KB
  crf_forward_kernel<<<Bn, 512, lds_bytes, stream>>>(x, trans, out_mx, out_pre,
                                                     lse_ws);

  finalize_kernel<<<1, 256, 0, stream>>>(trans, ws_x, lse_ws, out);
}